// LogLinearCDE_74474732913116
// MI455X (gfx1250) — compile-verified
//
#include <hip/hip_runtime.h>

// ---------------------------------------------------------------------------
// LogLinearCDE collapsed to:  P[h] = prod_t (1 + logsigs[t,:] . vf_A[:,h])
// GEMM part runs on V_WMMA_F32_16X16X4_F32 (fp32 matrix pipe, K=17 -> 5x K=4).
// Cross-half product combine hoisted out of the hot loop (products commute),
// so the loop body is pure VMEM + WMMA + VALU with no LDS waits.
// ---------------------------------------------------------------------------

typedef __attribute__((ext_vector_type(2))) float v2f;
typedef __attribute__((ext_vector_type(8))) float v8f;

#define LL      16384
#define HH      4096
#define CC      17
#define DD      16
#define LABELS  10
#define NCHUNK  32
#define CHUNK_STEPS (LL / NCHUNK)   // 512 timesteps per chunk
#define TILES_N (HH / 16)           // 256 hidden-column tiles

// ---------------------------------------------------------------------------
// Phase 1: each wave owns (16-column tile n0, time chunk) and computes the
// partial product over its 512 timesteps via fp32 WMMA.
//
// Layouts (CDNA5 ISA 7.12.2, 32-bit matrices, wave32):
//   A (16x4, MxK):  lanes 0-15: M=lane, {v0,v1}={K0,K1}; lanes 16-31: {K2,K3}
//   B (4x16, KxN):  lanes 0-15: N=lane, {v0,v1}={K0,K1}; lanes 16-31: {K2,K3}
//   C/D (16x16):    VGPR r: lanes 0-15 -> (M=r, N=lane); lanes 16-31 -> (M=r+8)
// ---------------------------------------------------------------------------
__global__ __launch_bounds__(32)
void loglincde_phase1(const float* __restrict__ logsigs,  // (L, C)
                      const float* __restrict__ vfA,      // (C, H)
                      float* __restrict__ partial)        // (NCHUNK, H)
{
    const int lane  = threadIdx.x;      // 0..31, full wave
    const int half  = lane >> 4;        // 0: K pair {0,1}; 1: K pair {2,3}
    const int m     = lane & 15;        // row (A) / column (B,D) index
    const int n0    = blockIdx.x * 16;  // hidden column tile base
    const int chunk = blockIdx.y;

    // Preload B tile: vf_A rows 0..16 (zero-pad K to 20), columns n0..n0+15.
    v2f B[5];
#pragma unroll
    for (int s = 0; s < 5; ++s) {
        const int k0 = 4 * s + 2 * half;
        float bx = (k0     < CC) ? vfA[(size_t)k0       * HH + n0 + m] : 0.0f;
        float by = (k0 + 1 < CC) ? vfA[(size_t)(k0 + 1) * HH + n0 + m] : 0.0f;
        v2f b; b.x = bx; b.y = by; B[s] = b;
    }

    // Running product of THIS lane-half's 8 rows per tile; halves are fused
    // with a single cross-lane shuffle after the loop (product commutes).
    float running = 1.0f;
    const int t_base = chunk * CHUNK_STEPS;

    for (int it = 0; it < CHUNK_STEPS / 16; ++it) {
        const int t0 = t_base + it * 16;

        // Lane (m, and its mirror m+16) loads logsig row t0+m: 17 coeffs.
        float a_row[20];
        const float* row = logsigs + (size_t)(t0 + m) * CC;
#pragma unroll
        for (int c = 0; c < CC; ++c) a_row[c] = row[c];
        a_row[17] = 0.0f; a_row[18] = 0.0f; a_row[19] = 0.0f;

        v8f acc = {};   // 16x16 tile of logsigs @ vf_A
#pragma unroll
        for (int s = 0; s < 5; ++s) {
            // constant indices + cndmask select (no dynamic register indexing)
            const float a0 = a_row[4 * s + 0], a1 = a_row[4 * s + 1];
            const float a2 = a_row[4 * s + 2], a3 = a_row[4 * s + 3];
            v2f a; a.x = half ? a2 : a0; a.y = half ? a3 : a1;
            acc = __builtin_amdgcn_wmma_f32_16x16x4_f32(
                    /*neg_a=*/false, a, /*neg_b=*/false, B[s],
                    /*c_mod=*/(short)0, acc, /*reuse_a=*/false, /*reuse_b=*/false);
        }

        // flows = 1 + acc; multiply this half's 8 time rows into the running
        // product (pairwise tree for ILP; compiler dual-issues these).
        float p01 = (1.0f + acc[0]) * (1.0f + acc[1]);
        float p23 = (1.0f + acc[2]) * (1.0f + acc[3]);
        float p45 = (1.0f + acc[4]) * (1.0f + acc[5]);
        float p67 = (1.0f + acc[6]) * (1.0f + acc[7]);
        running *= (p01 * p23) * (p45 * p67);
    }

    // Single cross-half combine: rows 0-7 (lanes 0-15) x rows 8-15 (lanes 16-31).
    running *= __shfl_xor(running, 16, 32);

    if (half == 0) partial[(size_t)chunk * HH + n0 + m] = running;
}

// ---------------------------------------------------------------------------
// Phase 2: reduce chunk partials, apply init layer, output head + softmax.
// Single block, fixed-order reductions -> bit-deterministic.
// ---------------------------------------------------------------------------
__global__ __launch_bounds__(1024)
void loglincde_phase2(const float* __restrict__ partial,  // (NCHUNK, H)
                      const float* __restrict__ x0,       // (D,)
                      const float* __restrict__ W_in,     // (H, D)
                      const float* __restrict__ b_in,     // (H,)
                      const float* __restrict__ W_out,    // (LABELS, H)
                      const float* __restrict__ b_out,    // (LABELS,)
                      float* __restrict__ out)            // (LABELS,)
{
    __shared__ float wsum[32 * LABELS];   // per-wave partial label sums
    const int tid  = threadIdx.x;
    const int lane = tid & 31;
    const int wid  = tid >> 5;            // 0..31 (wave32)

    float lsum[LABELS];
#pragma unroll
    for (int l = 0; l < LABELS; ++l) lsum[l] = 0.0f;

    for (int h = tid; h < HH; h += 1024) {
        float P = 1.0f;
#pragma unroll
        for (int c = 0; c < NCHUNK; ++c) P *= partial[(size_t)c * HH + h];
        float y0 = b_in[h];
#pragma unroll
        for (int d = 0; d < DD; ++d) y0 += W_in[(size_t)h * DD + d] * x0[d];
        const float yl = y0 * P;
#pragma unroll
        for (int l = 0; l < LABELS; ++l) lsum[l] += W_out[(size_t)l * HH + h] * yl;
    }

    // wave-level tree reduction (deterministic), leader stores to LDS
#pragma unroll
    for (int l = 0; l < LABELS; ++l) {
        float v = lsum[l];
#pragma unroll
        for (int off = 16; off > 0; off >>= 1) v += __shfl_down(v, off, 32);
        if (lane == 0) wsum[wid * LABELS + l] = v;
    }
    __syncthreads();

    if (tid == 0) {
        float logits[LABELS];
        float mx = -3.0e38f;
#pragma unroll
        for (int l = 0; l < LABELS; ++l) {
            float s = 0.0f;
            for (int w = 0; w < 32; ++w) s += wsum[w * LABELS + l];  // fixed order
            logits[l] = s + b_out[l];
            mx = fmaxf(mx, logits[l]);
        }
        float sum = 0.0f;
#pragma unroll
        for (int l = 0; l < LABELS; ++l) { logits[l] = __expf(logits[l] - mx); sum += logits[l]; }
        const float inv = 1.0f / sum;
#pragma unroll
        for (int l = 0; l < LABELS; ++l) out[l] = logits[l] * inv;
    }
}

// ---------------------------------------------------------------------------
// Inputs (setup_inputs order): 0 ts, 1 logsigs, 2 x0, 3 W_in, 4 b_in,
//                              5 vf_A, 6 W_out, 7 b_out.   ts is unused.
// ---------------------------------------------------------------------------
extern "C" void kernel_launch(void* const* d_in, const int* in_sizes, int n_in,
                              void* d_out, int out_size, void* d_ws, size_t ws_size,
                              hipStream_t stream) {
    const float* logsigs = (const float*)d_in[1];
    const float* x0      = (const float*)d_in[2];
    const float* W_in    = (const float*)d_in[3];
    const float* b_in    = (const float*)d_in[4];
    const float* vfA     = (const float*)d_in[5];
    const float* W_out   = (const float*)d_in[6];
    const float* b_out   = (const float*)d_in[7];

    float* partial = (float*)d_ws;   // NCHUNK * H floats = 512 KB scratch

    dim3 g1(TILES_N, NCHUNK);        // 256 x 32 waves
    loglincde_phase1<<<g1, 32, 0, stream>>>(logsigs, vfA, partial);
    loglincde_phase2<<<1, 1024, 0, stream>>>(partial, x0, W_in, b_in,
                                             W_out, b_out, (float*)d_out);
}